// GCN2_6158983102956
// MI455X (gfx1250) — compile-verified
//
#include <hip/hip_runtime.h>
#include <math.h>

typedef __attribute__((ext_vector_type(2))) float v2f;
typedef __attribute__((ext_vector_type(8))) float v8f;

// ---------------------------------------------------------------------------
// Degree / normalization
// ---------------------------------------------------------------------------
__global__ void k_init_deg(float* __restrict__ deg, int n) {
  int i = blockIdx.x * blockDim.x + threadIdx.x;
  if (i < n) deg[i] = 1.0f;  // self-loop contribution
}

__global__ void k_count_deg(const long long* __restrict__ dst, float* __restrict__ deg,
                            long long E) {
  long long e = (long long)blockIdx.x * blockDim.x + threadIdx.x;
  if (e < E) atomicAdd(&deg[dst[e]], 1.0f);
}

__global__ void k_dinv(const float* __restrict__ deg, float* __restrict__ dinv, int n) {
  int i = blockIdx.x * blockDim.x + threadIdx.x;
  if (i < n) {
    float d = deg[i];
    dinv[i] = (d > 0.0f) ? rsqrtf(d) : 0.0f;
  }
}

// ---------------------------------------------------------------------------
// FP32 WMMA GEMM: C[N,OUT] = X[N,K] @ W[K,OUT]  (optionally relu(X) on load)
// One wave -> 16 rows x (NTILES*16) cols, K-loop step 4 via v_wmma_f32_16x16x4_f32.
// A 16x4 layout: lanes 0-15 rows M, half0 holds K={k0,k0+1}, half1 K={k0+2,k0+3}.
// B 4x16 layout: VGPR0/1 = rows K (same half split), N striped on lanes 0-15.
// C/D 16x16: VGPR r -> M = r + 8*half, N = lane&15.
// ---------------------------------------------------------------------------
template <int K, int OUT, int NTILES, bool RELU_IN>
__global__ void k_gemm_wmma(const float* __restrict__ X, const float* __restrict__ W,
                            float* __restrict__ C, int N) {
  const int lane = threadIdx.x & 31;
  const int wave = threadIdx.x >> 5;
  const int half = lane >> 4;
  const int l    = lane & 15;
  const int row0 = (blockIdx.x * (blockDim.x >> 5) + wave) * 16;
  if (row0 >= N) return;  // wave-uniform exit: EXEC stays all-ones for WMMA

  v8f acc[NTILES];
  const v8f zero = {0.f, 0.f, 0.f, 0.f, 0.f, 0.f, 0.f, 0.f};
#pragma unroll
  for (int j = 0; j < NTILES; ++j) acc[j] = zero;

  int rowa = row0 + l;
  if (rowa >= N) rowa = N - 1;  // clamp (value-select only, no divergence at WMMA)
  const float* xrow = X + (size_t)rowa * K;

  for (int k0 = 0; k0 < K; k0 += 4) {
    float a0 = xrow[k0 + 2 * half + 0];
    float a1 = xrow[k0 + 2 * half + 1];
    if (RELU_IN) { a0 = fmaxf(a0, 0.0f); a1 = fmaxf(a1, 0.0f); }
    v2f a; a.x = a0; a.y = a1;
#pragma unroll
    for (int j = 0; j < NTILES; ++j) {
      const int col = j * 16 + l;
      v2f b;
      b.x = (col < OUT) ? W[(k0 + 2 * half + 0) * OUT + col] : 0.0f;
      b.y = (col < OUT) ? W[(k0 + 2 * half + 1) * OUT + col] : 0.0f;
      acc[j] = __builtin_amdgcn_wmma_f32_16x16x4_f32(
          /*neg_a=*/false, a, /*neg_b=*/false, b,
          /*c_mod=*/(short)0, acc[j], /*reuse_a=*/false, /*reuse_b=*/false);
    }
  }

  // Store epilogue: wave-uniform full-tile fast path (always taken when N%16==0)
  // -> 8 unguarded stores per tile at constant immediate offsets from one lane ptr.
  const bool fullRows = (row0 + 16 <= N);
  if (fullRows) {
#pragma unroll
    for (int j = 0; j < NTILES; ++j) {
      const int col = j * 16 + l;
      if (col < OUT) {
        float* cp = C + (size_t)(row0 + 8 * half) * OUT + col;
#pragma unroll
        for (int r = 0; r < 8; ++r) cp[(size_t)r * OUT] = acc[j][r];
      }
    }
  } else {
#pragma unroll
    for (int j = 0; j < NTILES; ++j) {
      const int col = j * 16 + l;
      if (col < OUT) {
        float* cp = C + (size_t)(row0 + 8 * half) * OUT + col;
#pragma unroll
        for (int r = 0; r < 8; ++r) {
          if (row0 + 8 * half + r < N) cp[(size_t)r * OUT] = acc[j][r];
        }
      }
    }
  }
}

// ---------------------------------------------------------------------------
// out[i,f] = h[i,f] * dinv[i]^2 + b[f]   (self-loop term + bias; full overwrite)
// ---------------------------------------------------------------------------
template <int F>
__global__ void k_selfloop_bias(const float* __restrict__ h, const float* __restrict__ dinv,
                                const float* __restrict__ b, float* __restrict__ out, int N) {
  long long t = (long long)blockIdx.x * blockDim.x + threadIdx.x;
  if (t >= (long long)N * F) return;
  int i = (int)(t / F);
  int f = (int)(t % F);
  float di = dinv[i];
  out[t] = h[t] * di * di + b[f];
}

// ---------------------------------------------------------------------------
// Edge scatter-add: one wave32 per edge. Uniform src/dst/dinv loads per wave,
// lane covers feature `lane` (and `lane+32`), coalesced gather + f32 atomics.
// ---------------------------------------------------------------------------
template <int F>
__global__ void k_edge_agg(const long long* __restrict__ src, const long long* __restrict__ dst,
                           const float* __restrict__ dinv, const float* __restrict__ h,
                           float* __restrict__ out, long long E) {
  long long e = (long long)blockIdx.x * (blockDim.x >> 5) + (threadIdx.x >> 5);
  if (e >= E) return;
  const int lane = threadIdx.x & 31;
  const long long s = src[e];
  const long long d = dst[e];
  const float w = dinv[s] * dinv[d];
  const float* hs = h + (size_t)s * F;
  float* od = out + (size_t)d * F;
  if (lane < F) atomicAdd(&od[lane], hs[lane] * w);
  if (F > 32 && lane + 32 < F) atomicAdd(&od[lane + 32], hs[lane + 32] * w);
}

// ---------------------------------------------------------------------------
// out = log_softmax(softmax(h)) per row of 40; one wave per row.
// ---------------------------------------------------------------------------
__global__ void k_softmax_log(const float* __restrict__ in, float* __restrict__ out, int N) {
  int row = blockIdx.x * (blockDim.x >> 5) + (threadIdx.x >> 5);
  if (row >= N) return;
  const int lane = threadIdx.x & 31;
  const int F = 40;
  const float* r = in + (size_t)row * F;
  float v0 = r[lane];                                        // lane < 32 < 40: valid
  float v1 = (lane + 32 < F) ? r[lane + 32] : -INFINITY;     // lanes 0..7

  float m = fmaxf(v0, v1);
#pragma unroll
  for (int off = 16; off > 0; off >>= 1) m = fmaxf(m, __shfl_xor(m, off, 32));
  float e0 = __expf(v0 - m);
  float e1 = (lane + 32 < F) ? __expf(v1 - m) : 0.0f;
  float s = e0 + e1;
#pragma unroll
  for (int off = 16; off > 0; off >>= 1) s += __shfl_xor(s, off, 32);
  float p0 = e0 / s;                // softmax
  float p1 = e1 / s;

  // log_softmax applied to probabilities p
  float q1 = (lane + 32 < F) ? p1 : -INFINITY;
  float m2 = fmaxf(p0, q1);
#pragma unroll
  for (int off = 16; off > 0; off >>= 1) m2 = fmaxf(m2, __shfl_xor(m2, off, 32));
  float g0 = __expf(p0 - m2);
  float g1 = (lane + 32 < F) ? __expf(p1 - m2) : 0.0f;
  float s2 = g0 + g1;
#pragma unroll
  for (int off = 16; off > 0; off >>= 1) s2 += __shfl_xor(s2, off, 32);
  float lse = m2 + __logf(s2);

  float* o = out + (size_t)row * F;
  o[lane] = p0 - lse;
  if (lane + 32 < F) o[lane + 32] = p1 - lse;
}

// ---------------------------------------------------------------------------
// Host-side orchestration
// ---------------------------------------------------------------------------
extern "C" void kernel_launch(void* const* d_in, const int* in_sizes, int n_in,
                              void* d_out, int out_size, void* d_ws, size_t ws_size,
                              hipStream_t stream) {
  const float* x  = (const float*)d_in[0];          // [N,128]
  const long long* ei = (const long long*)d_in[1];  // [2,E] int64
  const float* W1 = (const float*)d_in[2];          // [128,64]
  const float* b1 = (const float*)d_in[3];          // [64]
  const float* W2 = (const float*)d_in[4];          // [64,40]
  const float* b2 = (const float*)d_in[5];          // [40]
  float* out = (float*)d_out;                       // [N,40]

  const int N = in_sizes[0] / 128;
  const long long E = (long long)in_sizes[1] / 2;
  const long long* src = ei;
  const long long* dst = ei + E;

  // workspace layout (256B aligned), with aliasing:
  //   bufA: h0 [N,64]  -> reused as g2 [N,40]
  //   bufB: h1 [N,64]  -> reused as out_pre [N,40]
  char* ws = (char*)d_ws;
  size_t off = 0;
  auto carve = [&](size_t bytes) { size_t o = off; off = (off + bytes + 255) & ~(size_t)255; return o; };
  float* deg   = (float*)(ws + carve((size_t)N * 4));
  float* dinv  = (float*)(ws + carve((size_t)N * 4));
  float* bufA  = (float*)(ws + carve((size_t)N * 64 * 4));
  float* bufB  = (float*)(ws + carve((size_t)N * 64 * 4));
  (void)ws_size; (void)n_in; (void)out_size;

  const int TB = 256;                 // 8 waves
  const int WPB = TB / 32;

  // 1. degrees + dinv
  k_init_deg<<<(N + TB - 1) / TB, TB, 0, stream>>>(deg, N);
  k_count_deg<<<(unsigned)((E + TB - 1) / TB), TB, 0, stream>>>(dst, deg, E);
  k_dinv<<<(N + TB - 1) / TB, TB, 0, stream>>>(deg, dinv, N);

  // 2. h0 = x @ W1   (fp32 WMMA; 16 rows/wave, 8 waves/block -> 128 rows/block)
  k_gemm_wmma<128, 64, 4, false><<<(N + 16 * WPB - 1) / (16 * WPB), TB, 0, stream>>>(x, W1, bufA, N);

  // 3. h1 = selfloop(h0) + b1, then scatter-add over edges
  {
    long long tot = (long long)N * 64;
    k_selfloop_bias<64><<<(unsigned)((tot + TB - 1) / TB), TB, 0, stream>>>(bufA, dinv, b1, bufB, N);
    k_edge_agg<64><<<(unsigned)((E + WPB - 1) / WPB), TB, 0, stream>>>(src, dst, dinv, bufA, bufB, E);
  }

  // 4. g2 = relu(h1) @ W2  (ReLU fused into A-loads; g2 aliases bufA)
  k_gemm_wmma<64, 40, 3, true><<<(N + 16 * WPB - 1) / (16 * WPB), TB, 0, stream>>>(bufB, W2, bufA, N);

  // 5. out_pre = selfloop(g2) + b2, then scatter-add  (out_pre aliases bufB)
  {
    long long tot = (long long)N * 40;
    k_selfloop_bias<40><<<(unsigned)((tot + TB - 1) / TB), TB, 0, stream>>>(bufA, dinv, b2, bufB, N);
    k_edge_agg<40><<<(unsigned)((E + WPB - 1) / WPB), TB, 0, stream>>>(src, dst, dinv, bufA, bufB, E);
  }

  // 6. out = log_softmax(softmax(out_pre))
  k_softmax_log<<<(N + WPB - 1) / WPB, TB, 0, stream>>>(bufB, out, N);
}